// PCA_43404939493449
// MI455X (gfx1250) — compile-verified
//
#include <hip/hip_runtime.h>
#include <hip/hip_bf16.h>
#include <math.h>

typedef __attribute__((ext_vector_type(2))) float v2f;
typedef __attribute__((ext_vector_type(8))) float v8f;

#define C_CH 64
#define HW 16384
#define NCHUNK 8
#define CHUNK_K (HW / NCHUNK)      // 2048
#define SUB_K 64
#define NSUB (CHUNK_K / SUB_K)     // 32
#define LDS_STRIDE 68              // 64+4: conflict-free for the WMMA lane read pattern

// ---------------------------------------------------------------------------
// Stage 1: per-(batch, K-chunk) partial Gram G = x x^T (64x64, f32 WMMA) and
// per-channel partial sums. A-tile (16x4) and B-tile (4x16) of the same data
// share one per-lane layout: lane L holds element (idx = L&15) at
// K = k2 + 2*(L>>4) .. +1  (first K-half in lanes 0-15, second in 16-31).
// ---------------------------------------------------------------------------
__global__ __launch_bounds__(128)
void gram_stage1(const float* __restrict__ x,
                 float* __restrict__ gram_part,
                 float* __restrict__ sum_part) {
    const int b     = blockIdx.x;   // 0..31
    const int chunk = blockIdx.y;   // 0..NCHUNK-1
    const int tid   = threadIdx.x;  // 0..127
    const int lane  = tid & 31;
    const int wave  = tid >> 5;     // row-tile owned by this wave

    __shared__ float tile[C_CH * LDS_STRIDE];
    __shared__ float csum[C_CH];

    if (tid < C_CH) csum[tid] = 0.0f;

    const float* xb = x + (size_t)b * C_CH * HW + (size_t)chunk * CHUNK_K;

    // Load mapping: fixed channel per thread so channel sums accumulate locally.
    const int   lch  = tid >> 1;            // channel 0..63
    const int   loff = (tid & 1) * 32;      // k-half within 64-wide subchunk
    const float* srcRow = xb + (size_t)lch * HW + loff;
    float*       dstRow = &tile[lch * LDS_STRIDE + loff];

    v8f acc[4] = {v8f{}, v8f{}, v8f{}, v8f{}};
    float tsum = 0.0f;

    const int am = lane & 15;
    const int ak = (lane >> 4) * 2;
    const float* r0 = &tile[( 0 + am) * LDS_STRIDE + ak];
    const float* r1 = &tile[(16 + am) * LDS_STRIDE + ak];
    const float* r2 = &tile[(32 + am) * LDS_STRIDE + ak];
    const float* r3 = &tile[(48 + am) * LDS_STRIDE + ak];
    const float* ra = &tile[(16 * wave + am) * LDS_STRIDE + ak];

    for (int s = 0; s < NSUB; ++s) {
        __syncthreads();   // previous compute done before overwriting tile
        #pragma unroll
        for (int j = 0; j < 8; ++j) {
            float4 v = *(const float4*)(srcRow + s * SUB_K + 4 * j);
            *(float4*)(dstRow + 4 * j) = v;
            tsum += v.x + v.y + v.z + v.w;
        }
        __syncthreads();

        #pragma unroll
        for (int k2 = 0; k2 < SUB_K; k2 += 4) {
            v2f a  = *(const v2f*)(ra + k2);
            v2f t0 = *(const v2f*)(r0 + k2);
            v2f t1 = *(const v2f*)(r1 + k2);
            v2f t2 = *(const v2f*)(r2 + k2);
            v2f t3 = *(const v2f*)(r3 + k2);
            acc[0] = __builtin_amdgcn_wmma_f32_16x16x4_f32(
                false, a, false, t0, (short)0, acc[0], false, false);
            acc[1] = __builtin_amdgcn_wmma_f32_16x16x4_f32(
                false, a, false, t1, (short)0, acc[1], false, false);
            acc[2] = __builtin_amdgcn_wmma_f32_16x16x4_f32(
                false, a, false, t2, (short)0, acc[2], false, false);
            acc[3] = __builtin_amdgcn_wmma_f32_16x16x4_f32(
                false, a, false, t3, (short)0, acc[3], false, false);
        }
    }

    // channel partial sums (pairs of threads share a channel)
    atomicAdd(&csum[lch], tsum);

    // write partial Gram: C/D layout -> VGPR v holds M=v (lanes 0-15), M=v+8 (16-31)
    float* gp = gram_part + (size_t)(b * NCHUNK + chunk) * 4096;
    const int n_lo = lane & 15;
    const int m_hi = (lane >> 4) * 8;
    #pragma unroll
    for (int t = 0; t < 4; ++t) {
        #pragma unroll
        for (int v = 0; v < 8; ++v) {
            int m = 16 * wave + m_hi + v;
            gp[m * 64 + 16 * t + n_lo] = acc[t][v];
        }
    }

    __syncthreads();
    if (tid < C_CH) sum_part[(b * NCHUNK + chunk) * C_CH + tid] = csum[tid];
}

// ---------------------------------------------------------------------------
// Stage 2: c[b,c] = (sum_d (G - N*mu*mu^T)[c,d] * w1d[d]) / (N-1) + b1d
// ---------------------------------------------------------------------------
__global__ __launch_bounds__(64)
void cov_finalize(const float* __restrict__ gram_part,
                  const float* __restrict__ sum_part,
                  const float* __restrict__ w1d,
                  const float* __restrict__ b1d,
                  float* __restrict__ c_out) {
    const int b = blockIdx.x;
    const int c = threadIdx.x;
    __shared__ float mu[C_CH];
    __shared__ float wl[C_CH];

    float s = 0.0f;
    for (int k = 0; k < NCHUNK; ++k) s += sum_part[(b * NCHUNK + k) * C_CH + c];
    mu[c] = s * (1.0f / (float)HW);
    wl[c] = w1d[c];
    __syncthreads();

    float mudot = 0.0f;
    #pragma unroll
    for (int d = 0; d < C_CH; ++d) mudot += mu[d] * wl[d];

    float S = 0.0f;
    for (int k = 0; k < NCHUNK; ++k) {
        const float* g = gram_part + (size_t)(b * NCHUNK + k) * 4096 + c * 64;
        #pragma unroll 8
        for (int d = 0; d < C_CH; ++d) S = fmaf(g[d], wl[d], S);
    }
    float cw = (S - (float)HW * mu[c] * mudot) * (1.0f / (float)(HW - 1));
    c_out[b * C_CH + c] = cw + b1d[0];
}

// ---------------------------------------------------------------------------
// Depthwise 9x9 conv, padding 4. 32x32 output tile, 40x40 halo tile in LDS
// (row stride 41 -> conflict-free for the 4-per-lane column pattern).
// Each thread produces a 1x4 strip: 12 LDS loads feed 36 FMAs per tap row.
// ---------------------------------------------------------------------------
#define TILE 32
#define IN_T 40
#define IN_STRIDE 41

__global__ __launch_bounds__(256)
void dwconv9(const float* __restrict__ x,
             const float* __restrict__ wdw,
             const float* __restrict__ bdw,
             float* __restrict__ y) {
    __shared__ float tile[IN_T * IN_STRIDE];
    __shared__ float wl[81];

    const int bc  = blockIdx.y;               // b*64 + c
    const int c   = bc & 63;
    const int tx  = (blockIdx.x & 3) * TILE;
    const int ty  = (blockIdx.x >> 2) * TILE;
    const int tid = threadIdx.x;

    const float* xin = x + (size_t)bc * HW;

    if (tid < 81) wl[tid] = wdw[c * 81 + tid];
    for (int idx = tid; idx < IN_T * IN_T; idx += 256) {
        int r  = idx / IN_T;
        int cc = idx - r * IN_T;
        int gh = ty + r - 4;
        int gw = tx + cc - 4;
        float v = 0.0f;
        if ((unsigned)gh < 128u && (unsigned)gw < 128u) v = xin[gh * 128 + gw];
        tile[r * IN_STRIDE + cc] = v;
    }
    __syncthreads();

    const int r  = tid >> 3;          // 0..31
    const int c0 = (tid & 7) * 4;     // 0..28
    const float bias = bdw[c];
    float a0 = bias, a1 = bias, a2 = bias, a3 = bias;

    #pragma unroll
    for (int di = 0; di < 9; ++di) {
        const float* row = &tile[(r + di) * IN_STRIDE + c0];
        float v[12];
        #pragma unroll
        for (int j = 0; j < 12; ++j) v[j] = row[j];
        #pragma unroll
        for (int dj = 0; dj < 9; ++dj) {
            float w = wl[di * 9 + dj];
            a0 = fmaf(v[dj + 0], w, a0);
            a1 = fmaf(v[dj + 1], w, a1);
            a2 = fmaf(v[dj + 2], w, a2);
            a3 = fmaf(v[dj + 3], w, a3);
        }
    }
    float4 o = {a0, a1, a2, a3};
    *(float4*)&y[(size_t)bc * HW + (ty + r) * 128 + tx + c0] = o;
}

// ---------------------------------------------------------------------------
// att = sigmoid(c_y * (1 + softmax(c_y - c_x)))    per batch, 64 channels
// ---------------------------------------------------------------------------
__global__ __launch_bounds__(64)
void attention(const float* __restrict__ c_x, const float* __restrict__ c_y,
               float* __restrict__ att) {
    const int b = blockIdx.x;
    const int c = threadIdx.x;
    __shared__ float sd[C_CH];
    __shared__ float se[C_CH];

    float cy = c_y[b * C_CH + c];
    float d  = cy - c_x[b * C_CH + c];
    sd[c] = d;
    __syncthreads();
    float m = sd[0];
    #pragma unroll
    for (int i = 1; i < C_CH; ++i) m = fmaxf(m, sd[i]);
    float e = __expf(d - m);
    se[c] = e;
    __syncthreads();
    float sum = 0.0f;
    #pragma unroll
    for (int i = 0; i < C_CH; ++i) sum += se[i];
    float rz = e / sum;
    float z  = cy * (1.0f + rz);
    att[b * C_CH + c] = 1.0f / (1.0f + __expf(-z));
}

// ---------------------------------------------------------------------------
// out = y * att[b,c], float4 in place (y is recomputed each call -> idempotent)
// ---------------------------------------------------------------------------
__global__ __launch_bounds__(256)
void scale_out(float* __restrict__ y, const float* __restrict__ att) {
    size_t i4 = ((size_t)blockIdx.x * 256 + threadIdx.x) * 4;
    int chan  = (int)(i4 >> 14);          // b*64 + c  (16384 elems per channel)
    float a   = att[chan];
    float4 v  = *(float4*)&y[i4];
    v.x *= a; v.y *= a; v.z *= a; v.w *= a;
    *(float4*)&y[i4] = v;
}

// ---------------------------------------------------------------------------
extern "C" void kernel_launch(void* const* d_in, const int* in_sizes, int n_in,
                              void* d_out, int out_size, void* d_ws, size_t ws_size,
                              hipStream_t stream) {
    const float* x   = (const float*)d_in[0];
    const float* w1d = (const float*)d_in[1];
    const float* b1d = (const float*)d_in[2];
    const float* wdw = (const float*)d_in[3];
    const float* bdw = (const float*)d_in[4];
    float* y  = (float*)d_out;
    float* ws = (float*)d_ws;

    float* gram_part = ws;                                   // 32*8*4096 f
    float* sum_part  = gram_part + 32 * NCHUNK * 4096;       // 32*8*64 f
    float* c_x       = sum_part + 32 * NCHUNK * C_CH;        // 2048 f
    float* c_y       = c_x + 2048;                           // 2048 f
    float* att       = c_y + 2048;                           // 2048 f

    dim3 gGram(32, NCHUNK);

    // c_x from x
    gram_stage1<<<gGram, 128, 0, stream>>>(x, gram_part, sum_part);
    cov_finalize<<<32, 64, 0, stream>>>(gram_part, sum_part, w1d, b1d, c_x);

    // y = depthwise conv (staged in d_out)
    dwconv9<<<dim3(16, 32 * C_CH), 256, 0, stream>>>(x, wdw, bdw, y);

    // c_y from y
    gram_stage1<<<gGram, 128, 0, stream>>>(y, gram_part, sum_part);
    cov_finalize<<<32, 64, 0, stream>>>(gram_part, sum_part, w1d, b1d, c_y);

    // attention + in-place scale
    attention<<<32, 64, 0, stream>>>(c_x, c_y, att);
    scale_out<<<(32 * C_CH * HW) / (4 * 256), 256, 0, stream>>>(y, att);
}